// SMPLLayer_35510789603789
// MI455X (gfx1250) — compile-verified
//
#include <hip/hip_runtime.h>
#include <hip/hip_bf16.h>

typedef __attribute__((ext_vector_type(2))) float v2f;
typedef __attribute__((ext_vector_type(8))) float v8f;

#define BATCH 256
#define NV    6890
#define NJ    24
#define NB    10
#define NP    207
#define NPPAD 208
#define VC    (NV * 3)        // 20670
#define MT_TILES 1292         // ceil(20670/16)
#define MQ_BLOCKS (MT_TILES / 4)   // 323 blocks, 4 M-tiles each (64 rows)

// ---------------------------------------------------------------------------
// Kernel 1: Rodrigues -> R (B,J,9) and pose_coeffs (B,208) padded with zero
// ---------------------------------------------------------------------------
__global__ void k_rodrigues(const float* __restrict__ pose_axis,
                            float* __restrict__ Rws, float* __restrict__ pc) {
    int idx = blockIdx.x * 64 + threadIdx.x;
    if (idx >= BATCH * NJ) return;
    int b = idx / NJ, j = idx % NJ;
    float ax = pose_axis[b * (NJ * 3) + j * 3 + 0];
    float ay = pose_axis[b * (NJ * 3) + j * 3 + 1];
    float az = pose_axis[b * (NJ * 3) + j * 3 + 2];
    float n2 = ax * ax + ay * ay + az * az + 1e-16f;
    float ang = sqrtf(n2);
    float inv = 1.0f / ang;
    float kx = ax * inv, ky = ay * inv, kz = az * inv;
    float s = sinf(ang), c = cosf(ang), o = 1.0f - c;
    float R[9];
    R[0] = c + o * kx * kx;      R[1] = -s * kz + o * kx * ky; R[2] = s * ky + o * kx * kz;
    R[3] = s * kz + o * kx * ky; R[4] = c + o * ky * ky;       R[5] = -s * kx + o * ky * kz;
    R[6] = -s * ky + o * kx * kz;R[7] = s * kx + o * ky * kz;  R[8] = c + o * kz * kz;
#pragma unroll
    for (int e = 0; e < 9; e++) Rws[idx * 9 + e] = R[e];
    if (j > 0) {
#pragma unroll
        for (int e = 0; e < 9; e++)
            pc[b * NPPAD + (j - 1) * 9 + e] = R[e] - ((e % 4 == 0) ? 1.0f : 0.0f);
    } else {
        pc[b * NPPAD + NP] = 0.0f;  // K padding: makes B's K=207 row exactly zero
    }
}

// ---------------------------------------------------------------------------
// Kernel 2: v_shaped = v_template + shapedirs @ betas  (staged in v_tpose out)
// ---------------------------------------------------------------------------
__global__ void k_vshaped(const float* __restrict__ v_template,
                          const float* __restrict__ shapedirs,
                          const float* __restrict__ betas,
                          float* __restrict__ vsh_out) {
    int idx = blockIdx.x * 256 + threadIdx.x;      // exactly B*V threads
    int b = idx / NV, v = idx % NV;
    float bt[NB];
#pragma unroll
    for (int s = 0; s < NB; s++) bt[s] = betas[b * NB + s];
#pragma unroll
    for (int cc = 0; cc < 3; cc++) {
        float acc = v_template[v * 3 + cc];
        const float* sp = shapedirs + (v * 3 + cc) * NB;
#pragma unroll
        for (int s = 0; s < NB; s++) acc += sp[s] * bt[s];
        vsh_out[(size_t)(b * NV + v) * 3 + cc] = acc;
    }
}

// ---------------------------------------------------------------------------
// Kernel 3: j_rest[b][j][c] = sum_v J_regressor[j][v] * v_shaped[b][v][c]
// ---------------------------------------------------------------------------
__global__ void k_jrest(const float* __restrict__ Jreg,
                        const float* __restrict__ vsh,
                        float* __restrict__ jr) {
    __shared__ float red[3][128];
    int b = blockIdx.x / NJ, j = blockIdx.x % NJ;
    int tid = threadIdx.x;
    float a0 = 0.f, a1 = 0.f, a2 = 0.f;
    const float* jrow = Jreg + (size_t)j * NV;
    const float* vb = vsh + (size_t)b * NV * 3;
    for (int v = tid; v < NV; v += 128) {
        float w = jrow[v];
        a0 += w * vb[v * 3 + 0];
        a1 += w * vb[v * 3 + 1];
        a2 += w * vb[v * 3 + 2];
    }
    red[0][tid] = a0; red[1][tid] = a1; red[2][tid] = a2;
    __syncthreads();
    for (int s = 64; s > 0; s >>= 1) {
        if (tid < s) {
            red[0][tid] += red[0][tid + s];
            red[1][tid] += red[1][tid + s];
            red[2][tid] += red[2][tid + s];
        }
        __syncthreads();
    }
    if (tid == 0) {
        jr[(b * NJ + j) * 3 + 0] = red[0][0];
        jr[(b * NJ + j) * 3 + 1] = red[1][0];
        jr[(b * NJ + j) * 3 + 2] = red[2][0];
    }
}

// ---------------------------------------------------------------------------
// Kernel 4: WMMA f32 pose-blend GEMM with 4x4 register tiling.
//   Block = 4 waves; block covers 4 M-tiles (64 vc rows), wave w covers
//   N-tiles 4w..4w+3 (64 batches). Per K step: 4 A frags + 4 B frags ->
//   16 independent v_wmma_f32_16x16x4_f32 (loads:wmma = 1:2).
//   Guard-free K loop: 51 steps + tail (A garbage at K=207 annihilated by
//   zero-padded B row; OOB A rows only feed unstored D rows).
// ---------------------------------------------------------------------------
__global__ void __launch_bounds__(128)
k_vposed_wmma(const float* __restrict__ posedirs,
              const float* __restrict__ pc,
              float* __restrict__ vposed_out,
              float* __restrict__ vtpose_out /* holds v_shaped on entry */) {
    const int mtq  = blockIdx.x;         // M quad: rows mtq*64 .. mtq*64+63
    const int wv   = threadIdx.x >> 5;   // N quad: batches wv*64 .. wv*64+63
    const int lane = threadIdx.x & 31;
    const int half = lane >> 4;
    const int lr   = lane & 15;

    const float* __restrict__ arow[4];
#pragma unroll
    for (int i = 0; i < 4; i++) {
        int rowA = mtq * 64 + i * 16 + lr;
        int rowAc = rowA < VC ? rowA : VC - 1;       // clamp: safety only
        arow[i] = posedirs + (size_t)rowAc * NP;
    }
    const float* __restrict__ brow[4];
#pragma unroll
    for (int jj = 0; jj < 4; jj++)
        brow[jj] = pc + ((wv * 4 + jj) * 16 + lr) * NPPAD;

    v8f acc[4][4];
#pragma unroll
    for (int i = 0; i < 4; i++)
#pragma unroll
        for (int jj = 0; jj < 4; jj++)
            acc[i][jj] = (v8f){0.f,0.f,0.f,0.f,0.f,0.f,0.f,0.f};

    for (int k0 = 0; k0 < 51; k0++) {                // K = 0..203, no guards
        const int kb = k0 * 4 + half * 2;
        v2f a[4], b[4];
#pragma unroll
        for (int i = 0; i < 4; i++) { a[i].x = arow[i][kb]; a[i].y = arow[i][kb + 1]; }
#pragma unroll
        for (int jj = 0; jj < 4; jj++) { b[jj].x = brow[jj][kb]; b[jj].y = brow[jj][kb + 1]; }
#pragma unroll
        for (int i = 0; i < 4; i++)
#pragma unroll
            for (int jj = 0; jj < 4; jj++)
                acc[i][jj] = __builtin_amdgcn_wmma_f32_16x16x4_f32(
                    false, a[i], false, b[jj], (short)0, acc[i][jj], false, false);
    }
    {   // tail: K = 204..207 (207 zero-padded in B; A value there is don't-care)
        const int kb = 204 + half * 2;               // 204 or 206
        v2f a[4], b[4];
#pragma unroll
        for (int i = 0; i < 4; i++) { a[i].x = arow[i][kb]; a[i].y = arow[i][205 + half]; }
#pragma unroll
        for (int jj = 0; jj < 4; jj++) { b[jj].x = brow[jj][kb]; b[jj].y = brow[jj][kb + 1]; }
#pragma unroll
        for (int i = 0; i < 4; i++)
#pragma unroll
            for (int jj = 0; jj < 4; jj++)
                acc[i][jj] = __builtin_amdgcn_wmma_f32_16x16x4_f32(
                    false, a[i], false, b[jj], (short)0, acc[i][jj], false, false);
    }

    // D layout: VGPR r -> row M = half*8 + r, col N = lane&15 (batch)
#pragma unroll
    for (int jj = 0; jj < 4; jj++) {
        const int colB = (wv * 4 + jj) * 16 + lr;    // batch index
        float* vp = vposed_out + (size_t)colB * VC;
        float* vt = vtpose_out + (size_t)colB * VC;
#pragma unroll
        for (int i = 0; i < 4; i++) {
#pragma unroll
            for (int r = 0; r < 8; r++) {
                int vc = mtq * 64 + i * 16 + half * 8 + r;
                if (vc < VC) {
                    float val = vt[vc] + acc[i][jj][r];  // vt holds v_shaped
                    vp[vc] = val;
                    vt[vc] = val;                        // v_tpose == v_posed
                }
            }
        }
    }
}

// ---------------------------------------------------------------------------
// Kernel 5: kinematic chain per batch. 12 threads own the 3x4 of G.
//   Emits skinning matrices A (B,J,12) and joints output.
// ---------------------------------------------------------------------------
__global__ void k_chain(const float* __restrict__ Rws,
                        const float* __restrict__ jr,
                        const float* __restrict__ transl,
                        float* __restrict__ AM,
                        float* __restrict__ joints_out) {
    __shared__ float G[NJ][12];
    __shared__ float js[NJ][3];
    int b = blockIdx.x;
    int t = threadIdx.x;
    for (int i = t; i < NJ * 3; i += 32) js[i / 3][i % 3] = jr[b * NJ * 3 + i];
    __syncthreads();
    const int r = t >> 2, cc = t & 3;
    if (t < 12)
        G[0][t] = (cc < 3) ? Rws[(b * NJ + 0) * 9 + r * 3 + cc] : js[0][r];
    __syncthreads();
    const int parents[NJ] = {-1,0,0,0,1,2,3,4,5,6,7,8,9,9,9,12,13,14,16,17,18,19,20,21};
    for (int j = 1; j < NJ; j++) {
        int p = parents[j];
        if (t < 12) {
            float rel0 = (cc < 3) ? Rws[(b * NJ + j) * 9 + 0 * 3 + cc] : (js[j][0] - js[p][0]);
            float rel1 = (cc < 3) ? Rws[(b * NJ + j) * 9 + 1 * 3 + cc] : (js[j][1] - js[p][1]);
            float rel2 = (cc < 3) ? Rws[(b * NJ + j) * 9 + 2 * 3 + cc] : (js[j][2] - js[p][2]);
            float g = G[p][r * 4 + 0] * rel0 + G[p][r * 4 + 1] * rel1 + G[p][r * 4 + 2] * rel2;
            if (cc == 3) g += G[p][r * 4 + 3];
            G[j][t] = g;
        }
        __syncthreads();
    }
    for (int j = 0; j < NJ; j++) {
        if (t < 12) {
            float g = G[j][t];
            float out = g;
            if (cc == 3) {
                out = g - (G[j][r * 4 + 0] * js[j][0] +
                           G[j][r * 4 + 1] * js[j][1] +
                           G[j][r * 4 + 2] * js[j][2]);
                joints_out[(b * NJ + j) * 3 + r] = g + transl[b * 3 + r];
            }
            AM[(b * NJ + j) * 12 + t] = out;
        }
    }
}

// ---------------------------------------------------------------------------
// Kernel 6: LBS skinning. A_b cached in LDS; per-vertex 24-term blend.
// ---------------------------------------------------------------------------
__global__ void __launch_bounds__(256)
k_skin(const float* __restrict__ AM,
       const float* __restrict__ weights,
       const float* __restrict__ vposed,
       const float* __restrict__ transl,
       float* __restrict__ verts) {
    __shared__ float Am[NJ * 12];
    int b = blockIdx.y;
    int tid = threadIdx.x;
    for (int i = tid; i < NJ * 12; i += 256) Am[i] = AM[b * NJ * 12 + i];
    __syncthreads();
    int v = blockIdx.x * 256 + tid;
    if (v >= NV) return;
    const float* p = vposed + (size_t)(b * NV + v) * 3;
    float px = p[0], py = p[1], pz = p[2];
    float T[12];
#pragma unroll
    for (int e = 0; e < 12; e++) T[e] = 0.0f;
    const float* w = weights + (size_t)v * NJ;
    for (int j = 0; j < NJ; j++) {
        float wj = w[j];
#pragma unroll
        for (int e = 0; e < 12; e++) T[e] += wj * Am[j * 12 + e];
    }
    float x = T[0] * px + T[1] * py + T[2]  * pz + T[3];
    float y = T[4] * px + T[5] * py + T[6]  * pz + T[7];
    float z = T[8] * px + T[9] * py + T[10] * pz + T[11];
    float* o = verts + (size_t)(b * NV + v) * 3;
    o[0] = x + transl[b * 3 + 0];
    o[1] = y + transl[b * 3 + 1];
    o[2] = z + transl[b * 3 + 2];
}

// ---------------------------------------------------------------------------
extern "C" void kernel_launch(void* const* d_in, const int* in_sizes, int n_in,
                              void* d_out, int out_size, void* d_ws, size_t ws_size,
                              hipStream_t stream) {
    const float* pose_axis  = (const float*)d_in[0];
    const float* betas      = (const float*)d_in[1];
    const float* transl     = (const float*)d_in[2];
    const float* v_template = (const float*)d_in[3];
    const float* shapedirs  = (const float*)d_in[4];
    const float* posedirs   = (const float*)d_in[5];
    const float* Jreg       = (const float*)d_in[6];
    const float* weights    = (const float*)d_in[7];

    float* out    = (float*)d_out;
    float* verts  = out;                                  // B*V*3
    float* joints = out + (size_t)BATCH * NV * 3;         // B*J*3
    float* vposed = joints + (size_t)BATCH * NJ * 3;      // B*V*3
    float* vtpose = vposed + (size_t)BATCH * NV * 3;      // B*V*3 (scratch: v_shaped)

    float* ws  = (float*)d_ws;
    float* Rws = ws;                                      // B*J*9   = 55296
    float* pc  = Rws + BATCH * NJ * 9;                    // B*208   = 53248
    float* jr  = pc + BATCH * NPPAD;                      // B*J*3   = 18432
    float* AM  = jr + BATCH * NJ * 3;                     // B*J*12  = 73728

    // 1. Rodrigues + pose coefficients
    k_rodrigues<<<(BATCH * NJ + 63) / 64, 64, 0, stream>>>(pose_axis, Rws, pc);
    // 2. v_shaped (staged in v_tpose output region)
    k_vshaped<<<(BATCH * NV) / 256, 256, 0, stream>>>(v_template, shapedirs, betas, vtpose);
    // 3. rest joints
    k_jrest<<<BATCH * NJ, 128, 0, stream>>>(Jreg, vtpose, jr);
    // 4. pose blend GEMM via WMMA (writes v_posed and v_tpose)
    k_vposed_wmma<<<MQ_BLOCKS, 128, 0, stream>>>(posedirs, pc, vposed, vtpose);
    // 5. kinematic chain -> A matrices + joints
    k_chain<<<BATCH, 32, 0, stream>>>(Rws, jr, transl, AM, joints);
    // 6. skinning -> verts
    k_skin<<<dim3((NV + 255) / 256, BATCH), 256, 0, stream>>>(AM, weights, vposed, transl, verts);
}